// TropicalShortestPath_23295902613819
// MI455X (gfx1250) — compile-verified
//
#include <hip/hip_runtime.h>
#include <math.h>
#include <stdint.h>

typedef __attribute__((ext_vector_type(2))) float v2f;
typedef __attribute__((ext_vector_type(4))) float v4f;
typedef __attribute__((ext_vector_type(8))) float v8f;

#if defined(__has_builtin)
#if __has_builtin(__builtin_amdgcn_wmma_f32_16x16x4_f32)
#define HAVE_WMMA_F32X4 1
#endif
#endif

// Async-to-LDS staging via inline asm (builtin signature on this toolchain is
// descriptor-style; the ISA instruction itself is simple, so emit it directly).
#define USE_ASYNC_LDS 1

#define TILE_M 64
#define TILE_N 32
#define TILE_K 64
#define LDS_STRIDE 68   // padded row stride in floats: bank-conflict-free b64 frag reads

#define ACT_NONE 0
#define ACT_GELU 1
#define ACT_SIGMOID 2

__device__ __forceinline__ float act_apply(float x, int act) {
  if (act == ACT_GELU)    return 0.5f * x * (1.0f + erff(x * 0.70710678118654752f));
  if (act == ACT_SIGMOID) return 1.0f / (1.0f + __expf(-x));
  return x;
}

#if defined(USE_ASYNC_LDS)
// GLOBAL_LOAD_ASYNC_TO_LDS_B128 (VGLOBAL op 98): per-lane 16B memory->LDS,
// tracked by ASYNCcnt. VDST = per-lane LDS byte address (low 32 bits of the
// generic LDS pointer), VADDR = 64-bit global address, saddr=off.
__device__ __forceinline__ void async_b128(const float* g, float* l) {
  unsigned lds_addr = (unsigned)(uintptr_t)l;
  asm volatile("global_load_async_to_lds_b128 %0, %1, off"
               :: "v"(lds_addr), "v"(g) : "memory");
}
// Stage a rows x 64 (f32) tile: 16B per lane per instruction, uniform
// instruction count per thread (rows=64 -> 4, rows=32 -> 2).
__device__ __forceinline__ void async_stage(float* lds, const float* g, int ld,
                                            int rows, int tid) {
  for (int i = tid * 4; i < rows * TILE_K; i += 256 * 4) {
    int r = i / TILE_K, c = i % TILE_K;
    async_b128(g + (size_t)r * ld + c, lds + r * LDS_STRIDE + c);
  }
}
// SOPP immediate form confirmed by device disasm (s_wait_loadcnt 0x5 etc).
__device__ __forceinline__ void wait_async_0() {
  asm volatile("s_wait_asynccnt 0x0" ::: "memory");
}
__device__ __forceinline__ void wait_async_6() {
  asm volatile("s_wait_asynccnt 0x6" ::: "memory");
}
#endif

// C[M,N] = act(A[M,K] @ W[N,K]^T + bias[N])
// Block: 256 threads = 8 waves, wave grid 4x2 of 16x16 WMMA tiles -> 64x32 block tile.
// Double-buffered LDS tiles fed by GLOBAL_LOAD_ASYNC_TO_LDS_B128; the next tile
// stays in flight behind s_wait_asynccnt 6 while WMMA consumes the current one.
__global__ __launch_bounds__(256) void gemm_bias_act_wmma(
    const float* __restrict__ A, const float* __restrict__ W,
    const float* __restrict__ bias, float* __restrict__ C,
    int M, int N, int K, int act)
{
  const int tid   = threadIdx.x;
  const int lane  = tid & 31;
  const int wave  = tid >> 5;
  const int wrow  = wave >> 1;             // 0..3
  const int wcol  = wave & 1;              // 0..1

  const int m0 = blockIdx.y * TILE_M;
  const int n0 = blockIdx.x * TILE_N;

  const int mlane = lane & 15;
  const int hi    = lane >> 4;             // 0: lanes 0-15, 1: lanes 16-31
  const int koff  = hi * 2;                // K sub-offset per ISA A/B f32 layout

  const int colg = n0 + wcol * 16 + mlane; // C column for this lane

  v8f acc;
  {
    const float bv = bias[colg];
#pragma unroll
    for (int j = 0; j < 8; ++j) acc[j] = bv;
  }

#if defined(USE_ASYNC_LDS)
  __shared__ float As[2][TILE_M * LDS_STRIDE];
  __shared__ float Bs[2][TILE_N * LDS_STRIDE];

  const int nch = K / TILE_K;
  // Prologue: bring tile 0 in flight.
  async_stage(&As[0][0], A + (size_t)m0 * K, K, TILE_M, tid);
  async_stage(&Bs[0][0], W + (size_t)n0 * K, K, TILE_N, tid);

  for (int i = 0; i < nch; ++i) {
    const int cur = i & 1;
    if (i + 1 < nch) {
      const int k1 = (i + 1) * TILE_K;
      async_stage(&As[cur ^ 1][0], A + (size_t)m0 * K + k1, K, TILE_M, tid);
      async_stage(&Bs[cur ^ 1][0], W + (size_t)n0 * K + k1, K, TILE_N, tid);
      // 6 async ops/thread just issued for tile i+1; async loads complete
      // in order, so draining to <=6 outstanding means tile i fully landed.
      wait_async_6();
    } else {
      wait_async_0();
    }
    __syncthreads();

#if defined(HAVE_WMMA_F32X4)
    const float* ap = &As[cur][(wrow * 16 + mlane) * LDS_STRIDE + koff];
    const float* bp = &Bs[cur][(wcol * 16 + mlane) * LDS_STRIDE + koff];
#pragma unroll
    for (int kk = 0; kk < TILE_K; kk += 4) {
      v2f a = *(const v2f*)(ap + kk);
      v2f b = *(const v2f*)(bp + kk);
      acc = __builtin_amdgcn_wmma_f32_16x16x4_f32(
          false, a, false, b, (short)0, acc, false, false);
    }
#else
    const float* brow = &Bs[cur][(wcol * 16 + mlane) * LDS_STRIDE];
#pragma unroll
    for (int j = 0; j < 8; ++j) {
      const float* arow = &As[cur][(wrow * 16 + j + 8 * hi) * LDS_STRIDE];
      float s = 0.0f;
      for (int kk = 0; kk < TILE_K; ++kk) s += arow[kk] * brow[kk];
      acc[j] += s;
    }
#endif
    __syncthreads();  // all reads of buf[cur] done before it is refilled
  }

#else  // !USE_ASYNC_LDS: synchronous single-buffer staging

  __shared__ float As[TILE_M * LDS_STRIDE];
  __shared__ float Bs[TILE_N * LDS_STRIDE];

  for (int k0 = 0; k0 < K; k0 += TILE_K) {
    __syncthreads();
#pragma unroll
    for (int i = 0; i < (TILE_M * TILE_K) / (256 * 4); ++i) {
      int idx = (tid + i * 256) * 4;
      int r = idx / TILE_K, c = idx % TILE_K;
      *(v4f*)(&As[r * LDS_STRIDE + c]) =
          *(const v4f*)(A + (size_t)(m0 + r) * K + (k0 + c));
    }
#pragma unroll
    for (int i = 0; i < (TILE_N * TILE_K) / (256 * 4); ++i) {
      int idx = (tid + i * 256) * 4;
      int r = idx / TILE_K, c = idx % TILE_K;
      *(v4f*)(&Bs[r * LDS_STRIDE + c]) =
          *(const v4f*)(W + (size_t)(n0 + r) * K + (k0 + c));
    }
    __syncthreads();

#if defined(HAVE_WMMA_F32X4)
    const float* ap = &As[(wrow * 16 + mlane) * LDS_STRIDE + koff];
    const float* bp = &Bs[(wcol * 16 + mlane) * LDS_STRIDE + koff];
#pragma unroll
    for (int kk = 0; kk < TILE_K; kk += 4) {
      v2f a = *(const v2f*)(ap + kk);
      v2f b = *(const v2f*)(bp + kk);
      acc = __builtin_amdgcn_wmma_f32_16x16x4_f32(
          false, a, false, b, (short)0, acc, false, false);
    }
#else
    const float* brow = &Bs[(wcol * 16 + mlane) * LDS_STRIDE];
#pragma unroll
    for (int j = 0; j < 8; ++j) {
      const float* arow = &As[(wrow * 16 + j + 8 * hi) * LDS_STRIDE];
      float s = 0.0f;
      for (int kk = 0; kk < TILE_K; ++kk) s += arow[kk] * brow[kk];
      acc[j] += s;
    }
#endif
  }
#endif

  const int rbase = m0 + wrow * 16 + hi * 8; // C layout: VGPR j -> row j + 8*hi
#pragma unroll
  for (int j = 0; j < 8; ++j) {
    C[(size_t)(rbase + j) * N + colg] = act_apply(acc[j], act);
  }
}

// Row-wise LayerNorm over H=256; one block (256 threads) per row.
__global__ __launch_bounds__(256) void layernorm_256(
    const float* __restrict__ x, const float* __restrict__ g,
    const float* __restrict__ b, float* __restrict__ y)
{
  __shared__ float red[256];
  const int row = blockIdx.x;
  const int t = threadIdx.x;
  const float v = x[(size_t)row * 256 + t];

  red[t] = v; __syncthreads();
  for (int s = 128; s > 0; s >>= 1) { if (t < s) red[t] += red[t + s]; __syncthreads(); }
  const float mean = red[0] * (1.0f / 256.0f);
  __syncthreads();
  const float d = v - mean;
  red[t] = d * d; __syncthreads();
  for (int s = 128; s > 0; s >>= 1) { if (t < s) red[t] += red[t + s]; __syncthreads(); }
  const float var = red[0] * (1.0f / 256.0f);
  const float r = rsqrtf(var + 1e-5f);
  y[(size_t)row * 256 + t] = d * r * g[t] + b[t];
}

// Tropical min-plus (VALU min/add over LDS-staged W chunks) + LF dual activation
// + gated residual combine:  h = h + g*trop_out + (1-g)*cls
#define TROP_ROWS 16
#define TROP_OC   32
#define WPAD      260   // 260 % 64 == 4 -> at most 2-way bank conflict across o lanes

__global__ __launch_bounds__(256) void trop_lf_combine(
    const float* __restrict__ hn, float* __restrict__ h,
    const float* __restrict__ cls, const float* __restrict__ gate,
    const float* __restrict__ trop_W, const float* __restrict__ trop_b,
    const float* __restrict__ lfs_max, const float* __restrict__ lfi_max,
    const float* __restrict__ lfs_min, const float* __restrict__ lfi_min,
    const float* __restrict__ lf_mix)
{
  __shared__ float sh_hn[TROP_ROWS * 256];
  __shared__ float sh_w[TROP_OC * WPAD];

  const int tid = threadIdx.x;
  const int rb = blockIdx.x * TROP_ROWS;

  for (int i = tid * 4; i < TROP_ROWS * 256; i += 256 * 4) {
    int r = i / 256, c = i % 256;
    *(v4f*)(&sh_hn[i]) = *(const v4f*)(hn + (size_t)(rb + r) * 256 + c);
  }

  const int ol = tid & 31;   // o within chunk
  const int rg = tid >> 5;   // row group 0..7 -> rows 2*rg, 2*rg+1
  const int r0 = rg * 2, r1 = r0 + 1;

  for (int oc = 0; oc < 256; oc += TROP_OC) {
    __syncthreads();
    for (int i = tid * 4; i < TROP_OC * 256; i += 256 * 4) {
      int r = i / 256, c = i % 256;
      *(v4f*)(&sh_w[r * WPAD + c]) =
          *(const v4f*)(trop_W + (size_t)(oc + r) * 256 + c);
    }
    __syncthreads();

    const float* wr = &sh_w[ol * WPAD];
    const float* a0 = &sh_hn[r0 * 256];
    const float* a1 = &sh_hn[r1 * 256];
    float mn0 = 3.402823466e38f, mn1 = 3.402823466e38f;
#pragma unroll 8
    for (int i = 0; i < 256; ++i) {
      const float w = wr[i];
      mn0 = fminf(mn0, a0[i] + w);
      mn1 = fminf(mn1, a1[i] + w);
    }

    const int o = oc + ol;
    const float tb = trop_b[o];
    const float mix = 1.0f / (1.0f + __expf(-lf_mix[o]));
    float mns[2] = { mn0, mn1 };
    int   rr[2]  = { r0, r1 };
#pragma unroll
    for (int u = 0; u < 2; ++u) {
      const float t = mns[u] + tb;
      float mx = -3.402823466e38f, mnp = 3.402823466e38f;
#pragma unroll
      for (int p = 0; p < 8; ++p) {
        mx  = fmaxf(mx,  t * lfs_max[o * 8 + p] + lfi_max[o * 8 + p]);
        mnp = fminf(mnp, t * lfs_min[o * 8 + p] + lfi_min[o * 8 + p]);
      }
      const float tout = mix * mx + (1.0f - mix) * mnp;
      const size_t idx = (size_t)(rb + rr[u]) * 256 + o;
      const float gg = gate[idx];
      h[idx] = h[idx] + gg * tout + (1.0f - gg) * cls[idx];
    }
  }
}

extern "C" void kernel_launch(void* const* d_in, const int* in_sizes, int n_in,
                              void* d_out, int out_size, void* d_ws, size_t ws_size,
                              hipStream_t stream) {
  const float* adj     = (const float*)d_in[0];
  const float* ip_W    = (const float*)d_in[1];
  const float* ip_b    = (const float*)d_in[2];
  const float* norm_g  = (const float*)d_in[3];
  const float* norm_b  = (const float*)d_in[4];
  const float* trop_W  = (const float*)d_in[5];
  const float* trop_b  = (const float*)d_in[6];
  const float* lfs_max = (const float*)d_in[7];
  const float* lfi_max = (const float*)d_in[8];
  const float* lfs_min = (const float*)d_in[9];
  const float* lfi_min = (const float*)d_in[10];
  const float* lf_mix  = (const float*)d_in[11];
  const float* gate_W  = (const float*)d_in[12];
  const float* gate_b  = (const float*)d_in[13];
  const float* cls_W   = (const float*)d_in[14];
  const float* cls_b   = (const float*)d_in[15];
  const float* on_g    = (const float*)d_in[16];
  const float* on_b    = (const float*)d_in[17];
  const float* head_W  = (const float*)d_in[18];
  const float* head_b  = (const float*)d_in[19];
  float* out = (float*)d_out;

  const int NN = 4096;                       // N*N
  const int Bx = in_sizes[0] / NN;           // 4096
  const int H  = in_sizes[2];                // 256
  const int Lc = 4;

  float* ws   = (float*)d_ws;
  float* h    = ws;
  float* hn   = ws + (size_t)Bx * H;
  float* cls  = ws + 2 * (size_t)Bx * H;
  float* gate = ws + 3 * (size_t)Bx * H;

  dim3 blk(256);

  // Input projection: h = adj_flat @ ip_W^T + ip_b   (M=4096, N=256, K=4096)
  gemm_bias_act_wmma<<<dim3(H / TILE_N, Bx / TILE_M), blk, 0, stream>>>(
      adj, ip_W, ip_b, h, Bx, H, NN, ACT_NONE);

  for (int l = 0; l < Lc; ++l) {
    layernorm_256<<<dim3(Bx), blk, 0, stream>>>(h, norm_g + l * H, norm_b + l * H, hn);
    gemm_bias_act_wmma<<<dim3(H / TILE_N, Bx / TILE_M), blk, 0, stream>>>(
        hn, cls_W + (size_t)l * H * H, cls_b + l * H, cls, Bx, H, H, ACT_GELU);
    gemm_bias_act_wmma<<<dim3(H / TILE_N, Bx / TILE_M), blk, 0, stream>>>(
        hn, gate_W + (size_t)l * H * H, gate_b + l * H, gate, Bx, H, H, ACT_SIGMOID);
    trop_lf_combine<<<dim3(Bx / TROP_ROWS), blk, 0, stream>>>(
        hn, h, cls, gate,
        trop_W + (size_t)l * H * H, trop_b + l * H,
        lfs_max + (size_t)l * H * 8, lfi_max + (size_t)l * H * 8,
        lfs_min + (size_t)l * H * 8, lfi_min + (size_t)l * H * 8,
        lf_mix + l * H);
  }

  layernorm_256<<<dim3(Bx), blk, 0, stream>>>(h, on_g, on_b, hn);

  // Head: out = hn @ head_W^T + head_b   (M=4096, N=4096, K=256)
  gemm_bias_act_wmma<<<dim3(NN / TILE_N, Bx / TILE_M), blk, 0, stream>>>(
      hn, head_W, head_b, out, Bx, NN, H, ACT_NONE);
}